// NSHE_1554778161478
// MI455X (gfx1250) — compile-verified
//
#include <hip/hip_runtime.h>

// CDNA5 / gfx1250 (MI455X). wave32 only.
typedef __attribute__((ext_vector_type(2))) float v2f;
typedef __attribute__((ext_vector_type(8))) float v8f;

#define LDS_STRIDE 68   // 64 + 4 pad: float4-aligned, conflict-free fragment reads

// gfx1250 async global->LDS DMA (ASYNCcnt-tracked), per-lane 16B.
// VDST is the per-lane LDS byte address (flat addr truncated to 32b per ISA 10.2),
// VADDR the 64-bit global address.
__device__ __forceinline__ void async_copy_b128(void* lds_ptr, const void* gptr)
{
    unsigned lds_off = (unsigned)(uintptr_t)lds_ptr;
    unsigned long long gaddr = (unsigned long long)(uintptr_t)gptr;
    asm volatile("global_load_async_to_lds_b128 %0, %1, off"
                 :: "v"(lds_off), "v"(gaddr)
                 : "memory");
}

__device__ __forceinline__ void wait_async0()
{
    asm volatile("s_wait_asynccnt 0" ::: "memory");
}

// Fused: enc = relu(feat @ W + b); support = enc @ gcnW + gcnB
// Block = 256 threads = 8 wave32; block owns 32 rows.
// wave w: row-tile = w>>2 (16 rows), column slice = [(w&3)*16, +16).
__global__ __launch_bounds__(256) void encode_gcn_kernel(
    const float* __restrict__ feat, const float* __restrict__ W,
    const float* __restrict__ bias, const float* __restrict__ gcnW,
    const float* __restrict__ gcnB, float* __restrict__ support,
    int K, int rowOffset)
{
    __shared__ float lds_feat[32 * LDS_STRIDE];
    __shared__ float lds_enc [32 * LDS_STRIDE];

    const int lane  = threadIdx.x & 31;
    const int wave  = threadIdx.x >> 5;
    const int rt    = (wave >> 2) * 16;   // row-tile base within block
    const int n0    = (wave & 3) * 16;    // column-tile base
    const int lcol  = lane & 15;
    const int hi    = lane >> 4;
    const int khalf = hi * 2;             // K sub-offset for f32 fragments
    const int row0  = blockIdx.x * 32;

    // ---------------- GEMM1: 32x64 tile of relu(feat @ W + b) ----------------
    v8f acc;
    {
        const float b0 = bias[n0 + lcol];
        #pragma unroll
        for (int j = 0; j < 8; ++j) acc[j] = b0;
    }

    for (int kb = 0; kb < K; kb += 64) {
        // Async-stage a 32x64 chunk of features into LDS (ASYNCcnt path).
        #pragma unroll
        for (int i = 0; i < 2; ++i) {
            const int lin = threadIdx.x + i * 256;
            const int r   = lin >> 4;            // 0..31
            const int c4  = (lin & 15) << 2;     // 0,4,...,60
            async_copy_b128(&lds_feat[r * LDS_STRIDE + c4],
                            feat + (size_t)(row0 + r) * K + kb + c4);
        }
        if (kb + 64 < K) {
            // gfx1250 global_prefetch_b8 of next chunk
            __builtin_prefetch(feat + (size_t)(row0 + (threadIdx.x >> 3)) * K + kb + 64, 0, 0);
        }
        wait_async0();        // drain this wave's ASYNCcnt before signaling
        __syncthreads();

        #pragma unroll
        for (int k = 0; k < 64; k += 4) {
            const int kc = k + khalf;
            v2f a, b;
            // A 16x4 f32 fragment: lanes 0-15 -> K=kc,kc+1 ; lanes 16-31 -> K=kc+2,kc+3
            a.x = lds_feat[(rt + lcol) * LDS_STRIDE + kc];
            a.y = lds_feat[(rt + lcol) * LDS_STRIDE + kc + 1];
            // B 4x16 f32 fragment from row-major W[K][64] (L2-hot, shared by all blocks)
            const float* wp = W + (size_t)(kb + kc) * 64 + n0 + lcol;
            b.x = wp[0];
            b.y = wp[64];
            acc = __builtin_amdgcn_wmma_f32_16x16x4_f32(
                false, a, false, b, (short)0, acc, false, false);
        }
        __syncthreads();
    }

    // ReLU; stash enc tile in LDS for the fused GCN gemm.
    // D layout: VGPR j -> row (j + 8*hi), col (n0 + lcol)
    #pragma unroll
    for (int j = 0; j < 8; ++j) {
        lds_enc[(rt + j + hi * 8) * LDS_STRIDE + n0 + lcol] = fmaxf(acc[j], 0.0f);
    }
    __syncthreads();

    // ---------------- GEMM2: support = enc @ gcnW + gcnB (K = 64) ----------------
    v8f acc2;
    {
        const float b0 = gcnB[n0 + lcol];
        #pragma unroll
        for (int j = 0; j < 8; ++j) acc2[j] = b0;
    }
    #pragma unroll
    for (int k = 0; k < 64; k += 4) {
        const int kc = k + khalf;
        v2f a, b;
        a.x = lds_enc[(rt + lcol) * LDS_STRIDE + kc];
        a.y = lds_enc[(rt + lcol) * LDS_STRIDE + kc + 1];
        b.x = gcnW[kc * 64 + n0 + lcol];
        b.y = gcnW[(kc + 1) * 64 + n0 + lcol];
        acc2 = __builtin_amdgcn_wmma_f32_16x16x4_f32(
            false, a, false, b, (short)0, acc2, false, false);
    }

    // Store support tile (coalesced per VGPR row).
    const int col   = n0 + lcol;
    const int rbase = rowOffset + row0 + rt + hi * 8;
    #pragma unroll
    for (int j = 0; j < 8; ++j) {
        support[(size_t)(rbase + j) * 64 + col] = acc2[j];
    }
}

__global__ void zero_kernel(float* __restrict__ p, long long n)
{
    long long i = (long long)blockIdx.x * blockDim.x + threadIdx.x;
    if (i < n) p[i] = 0.0f;
}

// msg = support[src] * w ; out[dst] += msg.  One thread per (edge, column).
// support & out are L2-resident (51 MB each vs 192 MB L2).
__global__ __launch_bounds__(256) void spmm_kernel(
    const float* __restrict__ support, const float* __restrict__ ew,
    const int* __restrict__ esrc, const int* __restrict__ edst,
    float* __restrict__ out, int E)
{
    const int idx = blockIdx.x * 256 + threadIdx.x;
    const int e = idx >> 6;
    if (e >= E) return;
    const int d = idx & 63;
    const int s = esrc[e];
    const int t = edst[e];
    const float v = support[(size_t)s * 64 + d] * ew[e];
    atomicAdd(out + (size_t)t * 64 + d, v);
}

// Row-wise L2 normalize in place; one wave32 per 64-wide row.
__global__ __launch_bounds__(256) void normalize_kernel(float* __restrict__ out, int N)
{
    const int row = blockIdx.x * 8 + (threadIdx.x >> 5);
    if (row >= N) return;
    const int lane = threadIdx.x & 31;
    float* p = out + (size_t)row * 64;
    const float v0 = p[lane];
    const float v1 = p[lane + 32];
    float s = v0 * v0 + v1 * v1;
    #pragma unroll
    for (int off = 16; off > 0; off >>= 1) s += __shfl_xor(s, off, 32);
    const float inv = 1.0f / fmaxf(sqrtf(s), 1e-12f);
    p[lane]      = v0 * inv;
    p[lane + 32] = v1 * inv;
}

extern "C" void kernel_launch(void* const* d_in, const int* in_sizes, int n_in,
                              void* d_out, int out_size, void* d_ws, size_t ws_size,
                              hipStream_t stream)
{
    const float* feat_a = (const float*)d_in[0];
    const float* W_a    = (const float*)d_in[1];
    const float* b_a    = (const float*)d_in[2];
    const float* feat_b = (const float*)d_in[3];
    const float* W_b    = (const float*)d_in[4];
    const float* b_b    = (const float*)d_in[5];
    const float* feat_c = (const float*)d_in[6];
    const float* W_c    = (const float*)d_in[7];
    const float* b_c    = (const float*)d_in[8];
    const float* gcnW   = (const float*)d_in[9];
    const float* gcnB   = (const float*)d_in[10];
    const float* ew     = (const float*)d_in[11];
    const int*   esrc   = (const int*)d_in[12];
    const int*   edst   = (const int*)d_in[13];

    const int Ka = in_sizes[1] / 64;
    const int Kb = in_sizes[4] / 64;
    const int Kc = in_sizes[7] / 64;
    const int Na = in_sizes[0] / Ka;      // 100000 (multiple of 32)
    const int Nb = in_sizes[3] / Kb;      // 60000  (multiple of 32)
    const int Nc = in_sizes[6] / Kc;      // 40000  (multiple of 32)
    const int N  = Na + Nb + Nc;
    const int E  = in_sizes[11];

    float* support = (float*)d_ws;        // N*64 floats = 51.2 MB scratch
    float* out     = (float*)d_out;       // accumulate then normalize in place

    const long long nout = (long long)N * 64;
    zero_kernel<<<(int)((nout + 255) / 256), 256, 0, stream>>>(out, nout);

    encode_gcn_kernel<<<Na / 32, 256, 0, stream>>>(feat_a, W_a, b_a, gcnW, gcnB, support, Ka, 0);
    encode_gcn_kernel<<<Nb / 32, 256, 0, stream>>>(feat_b, W_b, b_b, gcnW, gcnB, support, Kb, Na);
    encode_gcn_kernel<<<Nc / 32, 256, 0, stream>>>(feat_c, W_c, b_c, gcnW, gcnB, support, Kc, Na + Nb);

    const long long work = (long long)E * 64;
    spmm_kernel<<<(int)((work + 255) / 256), 256, 0, stream>>>(support, ew, esrc, edst, out, E);

    normalize_kernel<<<(N + 7) / 8, 256, 0, stream>>>(out, N);
}